// NMSLayer_11879879543728
// MI455X (gfx1250) — compile-verified
//
#include <hip/hip_runtime.h>
#include <stdint.h>

#define B_ 8
#define N_ 10000
#define C_ 80
#define K_ 8
#define MAXT 8
#define IOU_THR_ 0.5f
#define SCORE_THR_ 0.5f
#define NEGV -1000000000.0f
#define TPB 256

#if defined(__clang__)
#if __has_builtin(__builtin_amdgcn_tensor_load_to_lds)
#define HAVE_TDM 1
#endif
#endif

typedef uint32_t u32x4 __attribute__((ext_vector_type(4)));
typedef int      i32x4 __attribute__((ext_vector_type(4)));
typedef int      i32x8 __attribute__((ext_vector_type(8)));

// ---------------------------------------------------------------------------
// Kernel 1: one workgroup per (batch, class).
// The strided score column (stride C*4 = 320B) is gathered into LDS by the
// Tensor Data Mover: one TENSOR_LOAD_TO_LDS with a 2-D descriptor
// (tile 1 x 10000, tensor_dim0_stride = 80, data_size = 4B) issued by wave 0,
// tracked by TENSORcnt. Fallback: per-lane GLOBAL_LOAD_ASYNC_TO_LDS_B32.
// Then K=8 serial rounds of block argmax + IoU suppression; boxes re-read
// from L2 as float4 (global_load_b128), 80x reuse per batch.
// ---------------------------------------------------------------------------
__global__ __launch_bounds__(TPB) void nms_per_class(
    const float* __restrict__ boxes,   // (B, N, 1, 4)
    const float* __restrict__ scores,  // (B, N, C)
    float* __restrict__ ws_sc,         // (B, C, K)    masked candidate scores
    float* __restrict__ ws_bx)         // (B, C, K, 4) candidate boxes (raw)
{
  const int c   = blockIdx.x;
  const int b   = blockIdx.y;
  const int tid = threadIdx.x;

  __shared__ float s[N_];     // 40000 B of 320KB/WGP
  __shared__ float rv[TPB];
  __shared__ int   ri[TPB];

  const float* colbase = scores + (size_t)b * N_ * C_ + c;

#if defined(HAVE_TDM)
  // ---- TDM: single-instruction strided gather scores[b, :, c] -> LDS ------
  if (tid < 32) {                       // wave 0 only (TDM ignores EXEC)
    const uint64_t ga   = (uint64_t)(const void*)colbase;
    // low 32 bits of a flat pointer to __shared__ == LDS byte offset
    const uint32_t lds0 = (uint32_t)(size_t)(&s[0]);

    u32x4 g0;
    g0[0] = 1u;                                    // count=1 (valid), no gather
    g0[1] = lds0;                                  // lds_addr [63:32]
    g0[2] = (uint32_t)ga;                          // global_addr [95:64]
    g0[3] = (uint32_t)((ga >> 32) & 0x01FFFFFFu)   // global_addr [120:96]
          | 0x80000000u;                           // type=2 @ bits [127:126]

    i32x8 g1;
    g1[0] = 0x00020000;            // data_size=2 (4B); mask/flags/pad = 0
    g1[1] = (int)(80u << 16);      // tensor_dim0[15:0]=80 @ bits 63:48
    g1[2] = (int)(10000u << 16);   // dim0 hi=0; tensor_dim1[15:0]=10000 @ 95:80
    g1[3] = (int)(1u << 16);       // dim1 hi=0; tile_dim0=1 @ bits 127:112
    g1[4] = 10000;                 // tile_dim1=10000 @ 143:128; tile_dim2=0
    g1[5] = 80;                    // tensor_dim0_stride[31:0]=80 @ 191:160
    g1[6] = 0;                     // stride hi=0; tensor_dim1_stride lo=0
    g1[7] = 0;

    i32x4 gz = {0, 0, 0, 0};       // groups 2/3 unused for 2-D tensor
#if __clang_major__ >= 23
    i32x8 g4 = {0, 0, 0, 0, 0, 0, 0, 0};
    __builtin_amdgcn_tensor_load_to_lds(g0, g1, gz, gz, g4, 0);
#else
    __builtin_amdgcn_tensor_load_to_lds(g0, g1, gz, gz, 0);
#endif
    __builtin_amdgcn_s_wait_tensorcnt(0);
  }
  __syncthreads();
#else
  // ---- fallback: per-lane async DMA gather (ASYNCcnt) ---------------------
  for (int j = tid; j < N_; j += TPB) {
    uint32_t ldsoff  = (uint32_t)(size_t)(&s[j]);
    const float* gp  = colbase + (size_t)j * C_;
    asm volatile("global_load_async_to_lds_b32 %0, %1, off"
                 :: "v"(ldsoff), "v"(gp) : "memory");
  }
  asm volatile("s_wait_asynccnt 0x0" ::: "memory");
  __syncthreads();
#endif

  // threshold pass
  for (int j = tid; j < N_; j += TPB) {
    float v = s[j];
    s[j] = (v > SCORE_THR_) ? v : NEGV;
  }
  __syncthreads();

  const float4* bx4 = (const float4*)boxes + (size_t)b * N_;
  float* osc = ws_sc + ((size_t)b * C_ + c) * K_;
  float* obx = ws_bx + ((size_t)b * C_ + c) * K_ * 4;

  for (int k = 0; k < K_; ++k) {
    // ---- block argmax, ties -> lowest index (matches jnp.argmax) ----------
    float bv = NEGV; int bi = 0;
    for (int j = tid; j < N_; j += TPB) {
      float v = s[j];
      if (v > bv) { bv = v; bi = j; }
    }
    rv[tid] = bv; ri[tid] = bi;
    __syncthreads();
    for (int off = TPB >> 1; off > 0; off >>= 1) {
      if (tid < off) {
        float v2 = rv[tid + off]; int i2 = ri[tid + off];
        if (v2 > rv[tid] || (v2 == rv[tid] && i2 < ri[tid])) {
          rv[tid] = v2; ri[tid] = i2;
        }
      }
      __syncthreads();
    }
    const float sc  = rv[0];
    const int   sel = ri[0];

    if (!(sc > SCORE_THR_)) {               // uniform: all remaining invalid
      if (tid == 0) {
        for (int kk = k; kk < K_; ++kk) {
          osc[kk] = NEGV;
          obx[kk * 4 + 0] = 0.f; obx[kk * 4 + 1] = 0.f;
          obx[kk * 4 + 2] = 0.f; obx[kk * 4 + 3] = 0.f;
        }
      }
      break;
    }

    const float4 bb = bx4[sel];             // uniform address -> broadcast
    if (tid == 0) {
      osc[k] = sc;
      obx[k * 4 + 0] = bb.x; obx[k * 4 + 1] = bb.y;
      obx[k * 4 + 2] = bb.z; obx[k * 4 + 3] = bb.w;
    }

    // ---- suppression sweep (L2-hot box reads, b128) -----------------------
    const float barea = (bb.z - bb.x) * (bb.w - bb.y);
    for (int j = tid; j < N_; j += TPB) {
      float v = s[j];
      if (v > 0.0f) {                       // survivors are > 0.5; rest NEG
        float4 q = bx4[j];
        float ih = fminf(bb.z, q.z) - fmaxf(bb.x, q.x); ih = fmaxf(ih, 0.0f);
        float iw = fminf(bb.w, q.w) - fmaxf(bb.y, q.y); iw = fmaxf(iw, 0.0f);
        float inter = ih * iw;
        float qarea = (q.z - q.x) * (q.w - q.y);
        float uni   = barea + qarea - inter;
        float iou   = inter / fmaxf(uni, 1e-9f);
        if (iou > IOU_THR_ || j == sel) s[j] = NEGV;
      }
    }
    __syncthreads();
  }
}

// ---------------------------------------------------------------------------
// Kernel 2: one workgroup per batch. top-8 over the C*K=640 masked candidate
// scores via argmax-with-removal (== lax.top_k: descending, ties lower idx).
// Output layout (floats, ints upcast): boxes[B,8,4] | scores[B,8] |
// classes[B,8] | valid_det[B]
// ---------------------------------------------------------------------------
__global__ __launch_bounds__(TPB) void topk_merge(
    const float* __restrict__ ws_sc,
    const float* __restrict__ ws_bx,
    float* __restrict__ out)
{
  const int b   = blockIdx.x;
  const int tid = threadIdx.x;
  const int M   = C_ * K_;                  // 640

  __shared__ float cs[C_ * K_];
  __shared__ float rv[TPB];
  __shared__ int   ri[TPB];

  for (int i = tid; i < M; i += TPB) cs[i] = ws_sc[(size_t)b * M + i];
  __syncthreads();

  float* obox = out;                           // B*8*4
  float* osco = out + B_ * MAXT * 4;           // B*8
  float* ocls = out + B_ * MAXT * 4 + B_ * MAXT;
  float* ovld = out + B_ * MAXT * 4 + 2 * B_ * MAXT;

  int cnt = 0;
  for (int m = 0; m < MAXT; ++m) {
    float bv = -3.0e38f; int bi = 0;
    for (int i = tid; i < M; i += TPB) {
      float v = cs[i];
      if (v > bv) { bv = v; bi = i; }
    }
    rv[tid] = bv; ri[tid] = bi;
    __syncthreads();
    for (int off = TPB >> 1; off > 0; off >>= 1) {
      if (tid < off) {
        float v2 = rv[tid + off]; int i2 = ri[tid + off];
        if (v2 > rv[tid] || (v2 == rv[tid] && i2 < ri[tid])) {
          rv[tid] = v2; ri[tid] = i2;
        }
      }
      __syncthreads();
    }
    const float sc = rv[0];
    const int   ii = ri[0];
    const bool valid = sc > SCORE_THR_;
    if (valid) cnt++;

    if (tid == 0) {
      float b0 = 0.f, b1 = 0.f, b2 = 0.f, b3 = 0.f, so = 0.f, cl = 0.f;
      if (valid) {
        const float* bp = ws_bx + ((size_t)b * M + ii) * 4;
        b0 = fminf(fmaxf(bp[0], 0.f), 1.f);
        b1 = fminf(fmaxf(bp[1], 0.f), 1.f);
        b2 = fminf(fmaxf(bp[2], 0.f), 1.f);
        b3 = fminf(fmaxf(bp[3], 0.f), 1.f);
        so = sc;
        cl = (float)(ii / K_);
      }
      const int o = (b * MAXT + m);
      obox[o * 4 + 0] = b0; obox[o * 4 + 1] = b1;
      obox[o * 4 + 2] = b2; obox[o * 4 + 3] = b3;
      osco[o] = so;
      ocls[o] = cl;
      cs[ii]  = -3.0e38f;                    // remove (below NEG)
    }
    __syncthreads();
  }
  if (tid == 0) ovld[b] = (float)cnt;
}

extern "C" void kernel_launch(void* const* d_in, const int* in_sizes, int n_in,
                              void* d_out, int out_size, void* d_ws, size_t ws_size,
                              hipStream_t stream) {
  (void)in_sizes; (void)n_in; (void)out_size; (void)ws_size;
  const float* boxes  = (const float*)d_in[0];   // (B, N, 1, 4)
  const float* scores = (const float*)d_in[1];   // (B, N, C)
  float* out   = (float*)d_out;
  float* ws_sc = (float*)d_ws;                           // B*C*K floats
  float* ws_bx = ws_sc + (size_t)B_ * C_ * K_;           // B*C*K*4 floats

  dim3 g1(C_, B_);
  nms_per_class<<<g1, TPB, 0, stream>>>(boxes, scores, ws_sc, ws_bx);
  topk_merge<<<B_, TPB, 0, stream>>>(ws_sc, ws_bx, out);
}